// SegmentationLoss_5703716569638
// MI455X (gfx1250) — compile-verified
//
#include <hip/hip_runtime.h>
#include <hip/hip_bf16.h>
#include <math.h>

#define B_   8
#define C_   20
#define HW_  (512*512)
#define CHW_ (C_*HW_)
#define TPB  256
#define BLOCKS_PER_IMG (HW_/TPB)   // 1024

// Workspace dword layout (zeroed by segloss_init):
//   f32 [0..7]   : per-batch BCE sums
//   f32 [8..15]  : per-batch focal sums
//   u32 [16..495]: per-batch histograms, 60 per batch: tgt[20], pred[20], tp[20]

__global__ void segloss_init(unsigned* ws) {
    int i = blockIdx.x * blockDim.x + threadIdx.x;
    if (i < 512) ws[i] = 0u;
}

__global__ __launch_bounds__(TPB) void segloss_main(
        const float* __restrict__ pred,
        const float* __restrict__ tgt,
        float* __restrict__ wsf,
        unsigned* __restrict__ wsu) {
    __shared__ float    tgt_tile[C_][TPB];   // 20 KB async-staged targets
    __shared__ unsigned hist[3 * C_];
    __shared__ float    wacc[16];

    const int tid = threadIdx.x;
    const int blk = blockIdx.x;
    const int b   = blk >> 10;                       // image index
    const int s   = ((blk & 1023) << 8) + tid;       // pixel within image

    if (tid < 3 * C_) hist[tid] = 0u;

    const size_t base = (size_t)b * CHW_ + (size_t)s;

    // ---- CDNA5 async-tensor path: prefetch one-hot targets into LDS (NT) ----
    {
        unsigned lds0 = (unsigned)(uintptr_t)(&tgt_tile[0][tid]); // low 32 bits = LDS byte offset
        unsigned long long ga = (unsigned long long)(uintptr_t)(tgt + base);
        #pragma unroll
        for (int c = 0; c < C_; ++c) {
            unsigned la = lds0 + (unsigned)(c * TPB * 4);
            unsigned long long g = ga + (unsigned long long)c * (unsigned long long)(HW_ * 4);
            asm volatile("global_load_async_to_lds_b32 %0, %1, off th:TH_LOAD_NT"
                         :: "v"(la), "v"(g) : "memory");
        }
    }

    // ---- stream predictions into registers (non-temporal: one-shot 168 MB) ----
    float x[C_];
    {
        const float* pb = pred + base;
        #pragma unroll
        for (int c = 0; c < C_; ++c)
            x[c] = __builtin_nontemporal_load(pb + (size_t)c * HW_);
    }

    // pass 1: max + argmax (prediction class)
    float m = x[0]; int P = 0;
    #pragma unroll
    for (int c = 1; c < C_; ++c) if (x[c] > m) { m = x[c]; P = c; }

    // pass 2: exponentials (cached in-place) + softmax denominator
    float Z = 0.0f;
    #pragma unroll
    for (int c = 0; c < C_; ++c) { x[c] = __expf(x[c] - m); Z += x[c]; }
    const float invZ = 1.0f / Z;

    // ---- wait for our own async LDS writes, then read label ----
    asm volatile("s_wait_asynccnt 0" ::: "memory");
    int L = 0;
    #pragma unroll
    for (int c = 0; c < C_; ++c) if (tgt_tile[c][tid] > 0.5f) L = c;

    // pass 3 (branch-free): sum log(1-p) over all classes, select p_L, patch label term
    float suml1 = 0.0f, pL = 0.0f;
    #pragma unroll
    for (int c = 0; c < C_; ++c) {
        const float p = x[c] * invZ;
        suml1 += fmaxf(__logf(1.0f - p), -100.0f);
        pL = (c == L) ? p : pL;
    }
    const float omL = 1.0f - pL;
    float bce = -suml1 + fmaxf(__logf(omL), -100.0f)   // remove label's log(1-p) term
                       - fmaxf(__logf(pL),  -100.0f);  // add -clip(log p_L, -100)
    float fcl = -__logf(pL + 1e-9f) * omL * omL;       // focal, gamma = 2

    __syncthreads();   // hist[] zeroing visible to all waves

    // confusion counts in LDS
    atomicAdd(&hist[L], 1u);
    atomicAdd(&hist[C_ + P], 1u);
    if (P == L) atomicAdd(&hist[2 * C_ + L], 1u);

    // wave32 reduction of bce / fcl
    #pragma unroll
    for (int off = 16; off > 0; off >>= 1) {
        bce += __shfl_xor(bce, off, 32);
        fcl += __shfl_xor(fcl, off, 32);
    }
    const int wave = tid >> 5;
    if ((tid & 31) == 0) { wacc[wave] = bce; wacc[8 + wave] = fcl; }
    __syncthreads();

    if (tid == 0) {
        float sb = 0.0f, sf = 0.0f;
        #pragma unroll
        for (int w = 0; w < 8; ++w) { sb += wacc[w]; sf += wacc[8 + w]; }
        atomicAdd(&wsf[b], sb);
        atomicAdd(&wsf[8 + b], sf);
    }
    if (tid < 3 * C_) atomicAdd(&wsu[b * (3 * C_) + tid], hist[tid]);
}

__global__ void segloss_final(const float* __restrict__ wsf,
                              const unsigned* __restrict__ wsu,
                              float* __restrict__ out) {
    if (threadIdx.x == 0 && blockIdx.x == 0) {
        float total = 0.0f;
        for (int b = 0; b < B_; ++b) {
            const float lseg = wsf[b]     * (1.0f / (float)CHW_);
            const float fcl  = wsf[8 + b] * (1.0f / (float)HW_);
            float ious = 0.0f, present = 0.0f;
            for (int c = 0; c < C_; ++c) {
                const float tg = (float)wsu[b * 60 + c];
                const float pr = (float)wsu[b * 60 + 20 + c];
                const float tp = (float)wsu[b * 60 + 40 + c];
                const float denom = tp + (pr - tp) + (tg - tp);
                if (denom > 0.0f) ious += tp / denom;
                if (tg > 0.0f)    present += 1.0f;
            }
            const float iou = ious / present;
            total += lseg + (1.0f - iou) + fcl;
        }
        out[0] = total * (1.0f / (float)B_);
    }
}

extern "C" void kernel_launch(void* const* d_in, const int* in_sizes, int n_in,
                              void* d_out, int out_size, void* d_ws, size_t ws_size,
                              hipStream_t stream) {
    const float* pred = (const float*)d_in[0];   // predictions [8,20,512,512] f32
    const float* tgt  = (const float*)d_in[1];   // one-hot targets, same shape
    float*    wsf = (float*)d_ws;
    unsigned* wsu = (unsigned*)d_ws + 16;
    float*    out = (float*)d_out;

    segloss_init<<<1, 512, 0, stream>>>((unsigned*)d_ws);
    segloss_main<<<B_ * BLOCKS_PER_IMG, TPB, 0, stream>>>(pred, tgt, wsf, wsu);
    segloss_final<<<1, 32, 0, stream>>>(wsf, wsu, out);
}